// Allocate_27676769255643
// MI455X (gfx1250) — compile-verified
//
#include <hip/hip_runtime.h>
#include <math.h>

typedef __attribute__((ext_vector_type(2))) float v2f;
typedef __attribute__((ext_vector_type(8))) float v8f;

#define DIMS 1024
#define M2   8192              // data2 rows
#define ROWS_PER_BLOCK 32      // 2 row-tiles of 16
#define NWAVES 8               // 2 row-tiles x 4 column-quarters
#define NCOLSPLIT 4
#define GROUP_COLS 64          // 4 WMMA fragments of 16 cols
#define NGROUPS (M2 / GROUP_COLS)              // 128
#define GROUPS_PER_SPLIT (NGROUPS / NCOLSPLIT) // 32
#define TOPK 8

// ---------------- row squared-norm kernel ----------------
__global__ __launch_bounds__(128) void row_norm2_kernel(
    const float* __restrict__ x, float* __restrict__ out, int nrows) {
  int row = blockIdx.x;
  if (row >= nrows) return;
  const float* p = x + (size_t)row * DIMS;
  float s = 0.f;
  for (int i = threadIdx.x; i < DIMS; i += blockDim.x) {
    float v = p[i];
    s += v * v;
  }
  // wave32 reduce
  for (int off = 16; off > 0; off >>= 1) s += __shfl_down(s, off, 32);
  __shared__ float partial[4];
  int wave = threadIdx.x >> 5;
  int lane = threadIdx.x & 31;
  if (lane == 0) partial[wave] = s;
  __syncthreads();
  if (threadIdx.x == 0) {
    out[row] = partial[0] + partial[1] + partial[2] + partial[3];
  }
}

// ---------------- fused GEMM + top-k + weighted gather ----------------
__global__ __launch_bounds__(256) void knn_wmma_kernel(
    const float* __restrict__ A,   // data1 [4096][1024]
    const float* __restrict__ B,   // data2 [8192][1024]
    const float* __restrict__ x2,  // [4096]
    const float* __restrict__ y2,  // [8192]
    float* __restrict__ out)       // [4096][1024]
{
  // +1 pad: row stride 65 floats -> 16 lanes reading a column hit 16 banks
  __shared__ float tile[NWAVES][16][GROUP_COLS + 1];          // ~33 KB
  __shared__ float cand_d[ROWS_PER_BLOCK][NCOLSPLIT * TOPK];  // 4 KB
  __shared__ int   cand_i[ROWS_PER_BLOCK][NCOLSPLIT * TOPK];  // 4 KB
  __shared__ float w_sh[ROWS_PER_BLOCK][TOPK];
  __shared__ int   i_sh[ROWS_PER_BLOCK][TOPK];
  __shared__ float wsum_sh[ROWS_PER_BLOCK];

  const int tid  = threadIdx.x;
  const int wave = tid >> 5;      // 0..7
  const int wr   = wave & 1;      // row-tile: waves with same wc share B lines
  const int wc   = wave >> 1;     // column-quarter: waves with same wr share A lines
  const int lane = tid & 31;
  const int half = lane >> 4;     // K-pair selector for f32 WMMA fragments
  const int l15  = lane & 15;
  const int r0   = blockIdx.x * ROWS_PER_BLOCK;
  const int myrow = r0 + wr * 16 + l15;

  // per-wave running top-8 of d^2 for myrow (meaningful in lanes 0..15)
  float topd[TOPK];
  int   topi[TOPK];
#pragma unroll
  for (int i = 0; i < TOPK; ++i) { topd[i] = 3.4e38f; topi[i] = 0; }
  float worst = 3.4e38f;
  int   worstslot = 0;

  const float x2r = x2[myrow];
  const float* Arow = A + (size_t)myrow * DIMS + 2 * half;

  const int g_begin = wc * GROUPS_PER_SPLIT;
  const int g_end   = g_begin + GROUPS_PER_SPLIT;

  for (int g = g_begin; g < g_end; ++g) {
    const int c0 = g * GROUP_COLS;
    v8f c0v = {}, c1v = {}, c2v = {}, c3v = {};
    const float* Brow = B + (size_t)(c0 + l15) * DIMS + 2 * half;

    for (int k = 0; k < DIMS; k += 4) {
      v2f a  = *(const v2f*)(Arow + k);
      v2f b0 = *(const v2f*)(Brow + k);
      v2f b1 = *(const v2f*)(Brow + (size_t)16 * DIMS + k);
      v2f b2 = *(const v2f*)(Brow + (size_t)32 * DIMS + k);
      v2f b3 = *(const v2f*)(Brow + (size_t)48 * DIMS + k);
      c0v = __builtin_amdgcn_wmma_f32_16x16x4_f32(false, a, false, b0,
                                                  (short)0, c0v, false, false);
      c1v = __builtin_amdgcn_wmma_f32_16x16x4_f32(false, a, false, b1,
                                                  (short)0, c1v, false, false);
      c2v = __builtin_amdgcn_wmma_f32_16x16x4_f32(false, a, false, b2,
                                                  (short)0, c2v, false, false);
      c3v = __builtin_amdgcn_wmma_f32_16x16x4_f32(false, a, false, b3,
                                                  (short)0, c3v, false, false);
    }

    // spill the 16x64 dot tile to LDS (C layout: VGPR v -> M = v + 8*half, N = l15)
#pragma unroll
    for (int v = 0; v < 8; ++v) {
      tile[wave][v + 8 * half][0 * 16 + l15] = c0v[v];
      tile[wave][v + 8 * half][1 * 16 + l15] = c1v[v];
      tile[wave][v + 8 * half][2 * 16 + l15] = c2v[v];
      tile[wave][v + 8 * half][3 * 16 + l15] = c3v[v];
    }
    // same-wave LDS RAW: DS ops are in-order per wave; fence compiler + wait
    asm volatile("s_wait_dscnt 0x0" ::: "memory");

    // lanes 0..15: scan 64 candidates for their row, keep smallest 8 d^2
    if (half == 0) {
      for (int j = 0; j < GROUP_COLS; ++j) {
        const int col = c0 + j;
        const float dot = tile[wave][l15][j];
        const float d2 = x2r + y2[col] - 2.0f * dot;
        if (d2 < worst) {
          topd[worstslot] = d2;
          topi[worstslot] = col;
          worst = topd[0]; worstslot = 0;
#pragma unroll
          for (int i = 1; i < TOPK; ++i)
            if (topd[i] > worst) { worst = topd[i]; worstslot = i; }
        }
      }
    }
    asm volatile("" ::: "memory");
  }

  // publish each wave's per-row top-8
  if (half == 0) {
#pragma unroll
    for (int i = 0; i < TOPK; ++i) {
      cand_d[wr * 16 + l15][wc * TOPK + i] = topd[i];
      cand_i[wr * 16 + l15][wc * TOPK + i] = topi[i];
    }
  }
  __syncthreads();

  // merge the 4 column-quarters' candidates (disjoint columns), build weights
  if (tid < ROWS_PER_BLOCK) {
    float td[TOPK]; int ti[TOPK];
#pragma unroll
    for (int i = 0; i < TOPK; ++i) { td[i] = 3.4e38f; ti[i] = 0; }
    float w2 = 3.4e38f; int ws = 0;
    for (int j = 0; j < NCOLSPLIT * TOPK; ++j) {
      const float d = cand_d[tid][j];
      if (d < w2) {
        td[ws] = d; ti[ws] = cand_i[tid][j];
        w2 = td[0]; ws = 0;
#pragma unroll
        for (int i = 1; i < TOPK; ++i)
          if (td[i] > w2) { w2 = td[i]; ws = i; }
      }
    }
    float wsum = 0.f;
#pragma unroll
    for (int i = 0; i < TOPK; ++i) {
      const float dist = sqrtf(fmaxf(td[i], 1e-12f));
      const float w = 1.0f / (dist + 0.1f);     // NUM / (d + CONST)
      w_sh[tid][i] = w;
      i_sh[tid][i] = ti[i];
      wsum += w;
    }
    wsum_sh[tid] = wsum;
  }
  __syncthreads();

  // weighted gather of the 8 neighbor rows, coalesced across 256 threads
  for (int r = 0; r < ROWS_PER_BLOCK; ++r) {
    float w[TOPK]; int ix[TOPK];
#pragma unroll
    for (int i = 0; i < TOPK; ++i) { w[i] = w_sh[r][i]; ix[i] = i_sh[r][i]; }
    const float inv = 1.0f / wsum_sh[r];
    for (int d = tid; d < DIMS; d += blockDim.x) {
      float acc = 0.f;
#pragma unroll
      for (int i = 0; i < TOPK; ++i)
        acc += w[i] * B[(size_t)ix[i] * DIMS + d];
      out[(size_t)(r0 + r) * DIMS + d] = acc * inv;
    }
  }
}

extern "C" void kernel_launch(void* const* d_in, const int* in_sizes, int n_in,
                              void* d_out, int out_size, void* d_ws, size_t ws_size,
                              hipStream_t stream) {
  const float* data1 = (const float*)d_in[0];  // [4096,1024] f32
  const float* data2 = (const float*)d_in[1];  // [8192,1024] f32
  const int N = in_sizes[0] / DIMS;            // 4096
  const int M = in_sizes[1] / DIMS;            // 8192

  float* x2 = (float*)d_ws;                    // N floats
  float* y2 = x2 + N;                          // M floats

  row_norm2_kernel<<<N, 128, 0, stream>>>(data1, x2, N);
  row_norm2_kernel<<<M, 128, 0, stream>>>(data2, y2, M);
  knn_wmma_kernel<<<N / ROWS_PER_BLOCK, 256, 0, stream>>>(data1, data2, x2, y2,
                                                          (float*)d_out);
}